// GraphSage_1735166787610
// MI455X (gfx1250) — compile-verified
//
#include <hip/hip_runtime.h>
#include <hip/hip_bf16.h>

typedef float v2f __attribute__((ext_vector_type(2)));
typedef float v4f __attribute__((ext_vector_type(4)));
typedef float v8f __attribute__((ext_vector_type(8)));

// ---------------------------------------------------------------------------
// Exclusive scan of (action[i] + 1) -> segment start offsets. Single block.
// offs[i] = i + sum_{j<i} action[j]; offs[n] = total neighbor count.
// ---------------------------------------------------------------------------
__global__ __launch_bounds__(1024) void scan_counts_kernel(
    const int* __restrict__ action, int n, int* __restrict__ offs) {
  __shared__ int smem[1024];
  const int tid = threadIdx.x;
  int carry = 0;  // replicated across all threads
  for (int base = 0; base < n; base += 1024) {
    int i = base + tid;
    int v = (i < n) ? (action[i] + 1) : 0;
    __syncthreads();  // protect smem reads from previous iteration
    smem[tid] = v;
    __syncthreads();
    for (int off = 1; off < 1024; off <<= 1) {
      int t = (tid >= off) ? smem[tid - off] : 0;
      __syncthreads();
      smem[tid] += t;
      __syncthreads();
    }
    int incl = smem[tid];
    int total = smem[1023];
    if (i < n) offs[i] = carry + incl - v;  // exclusive
    carry += total;
  }
  if (tid == 0) offs[n] = carry;
}

// ---------------------------------------------------------------------------
// Segment mean: one wave32 per segment, 4 floats/lane (D = 128).
// Rows are 512B -> one coalesced b128 per lane per neighbor row.
// This kernel carries the HBM roofline (h2 stream ~151 MB); prefetch the
// segment's cachelines up-front (global_prefetch_b8, no LOADcnt cost).
// ---------------------------------------------------------------------------
__global__ __launch_bounds__(256) void seg_mean_kernel(
    const float* __restrict__ neigh, const int* __restrict__ offs,
    int nseg, float* __restrict__ out) {
  const int seg = blockIdx.x * 8 + (threadIdx.x >> 5);
  const int lane = threadIdx.x & 31;
  if (seg >= nseg) return;
  const int s = offs[seg];
  const int e = offs[seg + 1];
  // Cover up to 8 KB (16 rows x 512 B) of the segment: 32 lanes x 256 B.
  __builtin_prefetch(neigh + (size_t)s * 128 + lane * 64, 0, 3);
  v4f acc = {0.f, 0.f, 0.f, 0.f};
  for (int j = s; j < e; ++j) {
    v4f v = *(const v4f*)(neigh + (size_t)j * 128 + lane * 4);
    acc += v;
  }
  const float inv = 1.0f / (float)(e - s);
  *(v4f*)(out + (size_t)seg * 128 + lane * 4) = acc * inv;
}

// ---------------------------------------------------------------------------
// Fused dual-input fp32 WMMA GEMM (compile-time K0/K1 and relu flags):
//   C[M x N] = act( A0[M x K0] @ Wa[K0 x N] + A1[M x K1] @ Wb[K1 x N] + bias )
// Block = 128 threads (4 waves). Each wave owns a 16x16 tile; block covers
// 64 rows x 16 cols. W column slab staged in LDS in B-fragment pair layout:
//   float2 slot (p, n) holds (W[2p][c0+n], W[2p+1][c0+n]).
// Per K-step(4): A frag = global b64, B frag = ds b64 (pairs merge into
// ds_load_2addr_b64), one v_wmma_f32_16x16x4_f32 (exact fp32 like reference).
// ---------------------------------------------------------------------------
template <int K0, int K1, bool RELU_IN, bool RELU_OUT>
__global__ __launch_bounds__(128) void gemm_wmma_f32_kernel(
    const float* __restrict__ A0, const float* __restrict__ Wa,
    const float* __restrict__ A1, const float* __restrict__ Wb,
    const float* __restrict__ bias, float* __restrict__ C, int M, int N) {
  constexpr int K = K0 + K1;
  __shared__ float ldsW[K * 16];
  const int tid = threadIdx.x;
  const int c0 = blockIdx.y * 16;

  // Stage W[:, c0:c0+16] into LDS (pair layout for the B fragment).
#pragma unroll
  for (int f = tid; f < K * 16; f += 128) {
    int k = f >> 4;
    int n = f & 15;
    float w = (k < K0) ? Wa[(size_t)k * N + c0 + n]
                       : Wb[(size_t)(k - K0) * N + c0 + n];
    ldsW[(((k >> 1) * 16) + n) * 2 + (k & 1)] = w;
  }
  __syncthreads();
  const v2f* ldsW2 = (const v2f*)ldsW;

  const int wave = tid >> 5;
  const int lane = tid & 31;
  const int half = lane >> 4;  // 0: K+0/K+1 ; 1: K+2/K+3 (ISA A/B striping)
  const int ln = lane & 15;

  const int row_base = blockIdx.x * 64 + wave * 16;
  int row = row_base + ln;
  int rowc = (row < M) ? row : (M - 1);  // clamp loads; EXEC stays all-1s
  const float* a0row = A0 + (size_t)rowc * K0;
  // Prefetch this wave's A0 row (K0*4 bytes) before the MAC stream.
#pragma unroll
  for (int p = 0; p < K0; p += 32) __builtin_prefetch(a0row + p, 0, 3);

  v8f acc = {};
#pragma unroll 8
  for (int k = 0; k < K0; k += 4) {
    v2f a = *(const v2f*)(a0row + k + half * 2);
    if (RELU_IN) {
      a.x = fmaxf(a.x, 0.f);
      a.y = fmaxf(a.y, 0.f);
    }
    v2f b = ldsW2[((k >> 1) + half) * 16 + ln];
    acc = __builtin_amdgcn_wmma_f32_16x16x4_f32(false, a, false, b, (short)0,
                                                acc, false, false);
  }
  if (K1 > 0) {
    const float* a1row = A1 + (size_t)rowc * K1;
#pragma unroll
    for (int p = 0; p < K1; p += 32) __builtin_prefetch(a1row + p, 0, 3);
#pragma unroll 8
    for (int k = 0; k < K1; k += 4) {
      v2f a = *(const v2f*)(a1row + k + half * 2);
      v2f b = ldsW2[(((K0 + k) >> 1) + half) * 16 + ln];
      acc = __builtin_amdgcn_wmma_f32_16x16x4_f32(false, a, false, b, (short)0,
                                                  acc, false, false);
    }
  }

  // Epilogue: C layout VGPR r -> M = r + 8*half, N = ln.
  float bv = bias ? bias[c0 + ln] : 0.0f;
#pragma unroll
  for (int r = 0; r < 8; ++r) {
    int orow = row_base + half * 8 + r;
    if (orow < M) {
      float v = acc[r] + bv;
      if (RELU_OUT) v = fmaxf(v, 0.f);
      C[(size_t)orow * N + c0 + ln] = v;
    }
  }
}

// ---------------------------------------------------------------------------
// Host-side orchestration.
// ---------------------------------------------------------------------------
static inline size_t align256(size_t x) { return (x + 255) & ~(size_t)255; }

extern "C" void kernel_launch(void* const* d_in, const int* in_sizes, int n_in,
                              void* d_out, int out_size, void* d_ws,
                              size_t ws_size, hipStream_t stream) {
  const int* action0 = (const int*)d_in[0];
  const int* action1 = (const int*)d_in[1];
  const float* h0 = (const float*)d_in[2];
  const float* h1 = (const float*)d_in[3];
  const float* h2 = (const float*)d_in[4];
  const float* W0m = (const float*)d_in[5];
  const float* B0m = (const float*)d_in[6];
  const float* W1m = (const float*)d_in[7];
  const float* B1m = (const float*)d_in[8];
  const float* fc1w = (const float*)d_in[9];
  const float* fc1b = (const float*)d_in[10];
  const float* fc2w = (const float*)d_in[11];
  const float* fc2b = (const float*)d_in[12];

  const int D = 128, HD = 256, OD = 64;
  const int N0 = in_sizes[2] / D;
  const int n1 = in_sizes[3] / D;

  // Carve workspace.
  char* ws = (char*)d_ws;
  size_t off = 0;
  int* offs0 = (int*)(ws + off); off = align256(off + (size_t)(N0 + 1) * 4);
  int* offs1 = (int*)(ws + off); off = align256(off + (size_t)(n1 + 1) * 4);
  float* m1  = (float*)(ws + off); off = align256(off + (size_t)n1 * D * 4);
  float* g1  = (float*)(ws + off); off = align256(off + (size_t)n1 * D * 4);
  float* m0  = (float*)(ws + off); off = align256(off + (size_t)N0 * D * 4);
  float* g0  = (float*)(ws + off); off = align256(off + (size_t)N0 * D * 4);
  float* mg1 = (float*)(ws + off); off = align256(off + (size_t)N0 * D * 4);
  float* x1  = (float*)(ws + off); off = align256(off + (size_t)N0 * HD * 4);

  float* out = (float*)d_out;            // [N0 x OD]
  float* f0 = out + (size_t)N0 * OD;     // [N0 x D]

  // 1) Segment offsets.
  scan_counts_kernel<<<1, 1024, 0, stream>>>(action0, N0, offs0);
  scan_counts_kernel<<<1, 1024, 0, stream>>>(action1, n1, offs1);

  // 2) Segment means of raw features.
  seg_mean_kernel<<<(n1 + 7) / 8, 256, 0, stream>>>(h2, offs1, n1, m1);
  seg_mean_kernel<<<(N0 + 7) / 8, 256, 0, stream>>>(h1, offs0, N0, m0);

  // 3) SAGE layer 0: g = relu(src @ B0 + mean @ W0), fused as K=256 GEMM.
  {
    dim3 grid((n1 + 63) / 64, D / 16);
    gemm_wmma_f32_kernel<128, 128, false, true>
        <<<grid, 128, 0, stream>>>(h1, B0m, m1, W0m, nullptr, g1, n1, D);
  }
  {
    dim3 grid((N0 + 63) / 64, D / 16);
    gemm_wmma_f32_kernel<128, 128, false, true>
        <<<grid, 128, 0, stream>>>(h0, B0m, m0, W0m, nullptr, g0, N0, D);
  }

  // 4) Mean of g1 over hop-0 segments.
  seg_mean_kernel<<<(N0 + 7) / 8, 256, 0, stream>>>(g1, offs0, N0, mg1);

  // 5) SAGE layer 1 (no activation): f0 = g0 @ B1 + mg1 @ W1.
  {
    dim3 grid((N0 + 63) / 64, D / 16);
    gemm_wmma_f32_kernel<128, 128, false, false>
        <<<grid, 128, 0, stream>>>(g0, B1m, mg1, W1m, nullptr, f0, N0, D);
  }

  // 6) Head: x1 = relu(relu(f0) @ fc1_w + fc1_b)  (RELU_IN fuses relu(f0)).
  {
    dim3 grid((N0 + 63) / 64, HD / 16);
    gemm_wmma_f32_kernel<128, 0, true, true><<<grid, 128, 0, stream>>>(
        f0, fc1w, nullptr, nullptr, fc1b, x1, N0, HD);
  }

  // 7) out = x1 @ fc2_w + fc2_b.
  {
    dim3 grid((N0 + 63) / 64, OD / 16);
    gemm_wmma_f32_kernel<128, 0, false, false><<<grid, 128, 0, stream>>>(
        x1, fc2w, nullptr, nullptr, fc2b, out, N0, OD);
  }
}